// MaskPred_4801773437472
// MI455X (gfx1250) — compile-verified
//
#include <hip/hip_runtime.h>
#include <hip/hip_bf16.h>

// CDNA5 / gfx1250, wave32. GEMM-reformulated EmbedMask mask prediction.
//
//   dist = [ M | -2*M.*E ] (256x64) @ [ P.^2 ; P ] (64x15200) + c[o]
//   out  = exp(-dist), row o taken from image im_inds[o].
//
// Main kernel: one wave32 per 16x16 output tile, 16x V_WMMA_F32_16X16X4_F32
// (K = 64 in steps of 4). Memory-bound problem (15.6 MB output @ 23.3 TB/s),
// f32 WMMA keeps full precision at zero roofline cost.

typedef float v2f __attribute__((ext_vector_type(2)));
typedef float v8f __attribute__((ext_vector_type(8)));

#define N_IM 2
#define DIMC 32
#define HW   (100 * 152)   // 15200, exactly 950 tiles of 16
#define OBJ  256
#define KTOT 64            // 2*DIMC

// ---------------------------------------------------------------------------
// Prep: build A-matrix (K-major: Amat[k][o], 64x256) and per-object constant
//   Amat[d][o]      =  margin[o][d]                (pairs with P^2 rows)
//   Amat[32+d][o]   = -2 * margin[o][d] * e[o][d]  (pairs with P rows)
//   c[o]            =  sum_d margin[o][d] * e[o][d]^2
// ---------------------------------------------------------------------------
__global__ void maskpred_prep(const float* __restrict__ embed,
                              const float* __restrict__ margin,
                              float* __restrict__ Amat,
                              float* __restrict__ cvec) {
    int o = blockIdx.x * blockDim.x + threadIdx.x;
    if (o >= OBJ) return;
    float c = 0.0f;
#pragma unroll
    for (int d = 0; d < DIMC; ++d) {
        float m = margin[o * DIMC + d];
        float e = embed[o * DIMC + d];
        Amat[d * OBJ + o]          = m;
        Amat[(DIMC + d) * OBJ + o] = -2.0f * m * e;
        c = fmaf(m, e * e, c);
    }
    cvec[o] = c;
}

// ---------------------------------------------------------------------------
// Main: grid = (950 pixel tiles, 16 object tiles, 2 images), block = 1 wave32.
// Each wave: load 8 raw B fragments (coalesced along pixels), square in-reg,
// run 16 WMMA steps, add c[o], exp, store rows whose im_inds matches img.
// ---------------------------------------------------------------------------
__global__ void __launch_bounds__(32)
maskpred_wmma(const float* __restrict__ pixel_embed,
              const int*   __restrict__ im_inds,
              const float* __restrict__ Amat,
              const float* __restrict__ cvec,
              float*       __restrict__ out) {
    const int pt    = blockIdx.x << 4;     // pixel tile base (N dim)
    const int obase = blockIdx.y << 4;     // object tile base (M dim)
    const int img   = blockIdx.z;          // which image's pixels
    const int lane  = threadIdx.x;         // 0..31
    const int half  = lane >> 4;           // 0: K pair {0,1}, 1: K pair {2,3}
    const int nl    = lane & 15;           // column / row-within-tile index

    // B fragments: lane holds B[kb + 2*half + {0,1}][pt + nl].
    // Row k<32 of B is pix[d]^2, row 32+k is pix[d]; same pixel values, so
    // load raw once and square in registers (pixel_embed read exactly once).
    const float* pbase = pixel_embed + (size_t)img * DIMC * HW + pt + nl;
    v2f braw[8], bsq[8];
#pragma unroll
    for (int j = 0; j < 8; ++j) {
        const int d = 4 * j + 2 * half;
        v2f b;
        b.x = pbase[(size_t)d * HW];        // 16-lane contiguous 64B segment
        b.y = pbase[(size_t)(d + 1) * HW];
        braw[j] = b;
        bsq[j].x = b.x * b.x;
        bsq[j].y = b.y * b.y;
    }

    v8f acc = {0.f, 0.f, 0.f, 0.f, 0.f, 0.f, 0.f, 0.f};
    const float* arow = Amat + obase + nl;  // A frag: lane-contiguous in o

    // K = 0..31 : margin x pix^2
#pragma unroll
    for (int j = 0; j < 8; ++j) {
        const int k = 4 * j + 2 * half;
        v2f a;
        a.x = arow[(size_t)k * OBJ];
        a.y = arow[(size_t)(k + 1) * OBJ];
        acc = __builtin_amdgcn_wmma_f32_16x16x4_f32(
            false, a, false, bsq[j], (short)0, acc, false, false);
    }
    // K = 32..63 : (-2*margin*embed) x pix
#pragma unroll
    for (int j = 0; j < 8; ++j) {
        const int k = 32 + 4 * j + 2 * half;
        v2f a;
        a.x = arow[(size_t)k * OBJ];
        a.y = arow[(size_t)(k + 1) * OBJ];
        acc = __builtin_amdgcn_wmma_f32_16x16x4_f32(
            false, a, false, braw[j], (short)0, acc, false, false);
    }

    // D layout: VGPR r, lane l -> M = r + 8*(l>=16), N = l&15.
    // Store only rows belonging to this image (each element written once).
#pragma unroll
    for (int r = 0; r < 8; ++r) {
        const int o = obase + r + 8 * half;
        if (im_inds[o] == img) {
            const float dist = acc[r] + cvec[o];
            out[(size_t)o * HW + pt + nl] = __expf(-dist);
        }
    }
}

extern "C" void kernel_launch(void* const* d_in, const int* in_sizes, int n_in,
                              void* d_out, int out_size, void* d_ws, size_t ws_size,
                              hipStream_t stream) {
    const float* pixel_embed = (const float*)d_in[0];   // [2,32,100,152]
    const float* prop_embed  = (const float*)d_in[1];   // [256,32]
    const float* prop_margin = (const float*)d_in[2];   // [256,32]
    const int*   im_inds     = (const int*)d_in[3];     // [256]
    // d_in[4] = mask_feat_stride (unused: upsample factor is 1)

    float* Amat = (float*)d_ws;            // 64*256 f32 = 64 KB
    float* cvec = Amat + KTOT * OBJ;       // 256 f32

    maskpred_prep<<<1, 256, 0, stream>>>(prop_embed, prop_margin, Amat, cvec);

    dim3 grid(HW / 16, OBJ / 16, N_IM);    // (950, 16, 2)
    maskpred_wmma<<<grid, 32, 0, stream>>>(pixel_embed, im_inds, Amat, cvec,
                                           (float*)d_out);
}